// e3jLayer_33612414059053
// MI455X (gfx1250) — compile-verified
//
#include <hip/hip_runtime.h>

typedef __attribute__((ext_vector_type(2))) float v2f;
typedef __attribute__((ext_vector_type(8))) float v8f;

#define INV_S3 0.57735026918962576451f
#define INV_S2 0.70710678118654752440f

// ---------------------------------------------------------------------------
// Kernel 1: zero the accumulator (d_out is used as the f32 accumulation buf)
// ---------------------------------------------------------------------------
__global__ void zero_f4_kernel(float4* __restrict__ out, int n4) {
    int i = blockIdx.x * blockDim.x + threadIdx.x;
    if (i < n4) out[i] = make_float4(0.f, 0.f, 0.f, 0.f);
}

// ---------------------------------------------------------------------------
// Kernel 2: per-(edge, feature) tensor product + scatter-add.
// 16 consecutive lanes handle one edge (one lane per feature f=0..15).
// ---------------------------------------------------------------------------
__device__ __forceinline__ void atom_add_f32(float* p, float v) {
    (void)__hip_atomic_fetch_add(p, v, __ATOMIC_RELAXED, __HIP_MEMORY_SCOPE_AGENT);
}

__global__ void edge_tp_scatter_kernel(const float* __restrict__ pos,
                                       const float* __restrict__ feats,
                                       const int*   __restrict__ senders,
                                       const int*   __restrict__ receivers,
                                       float*       __restrict__ out,
                                       int E) {
    int tid = blockIdx.x * blockDim.x + threadIdx.x;
    int e = tid >> 4;          // edge index
    int f = tid & 15;          // feature index
    if (e >= E) return;

    int s = senders[e];
    int r = receivers[e];

    // normalized edge direction (redundant across the 16 lanes of this edge;
    // positions are 1.2 MB -> L2/L0 resident, cheap)
    float dx = pos[r * 3 + 0] - pos[s * 3 + 0];
    float dy = pos[r * 3 + 1] - pos[s * 3 + 1];
    float dz = pos[r * 3 + 2] - pos[s * 3 + 2];
    float nrm = sqrtf(dx * dx + dy * dy + dz * dz);
    float il = 1.0f / (nrm + 1e-9f);
    dx *= il; dy *= il; dz *= il;

    // gather 4 irrep components for this feature: contiguous 16B
    const float4 x = *(const float4*)(feats + ((size_t)s * 16 + f) * 4);
    float x0 = x.x, x1 = x.y, x2 = x.z, x3 = x.w;

    float dot = x1 * dx + x2 * dy + x3 * dz;
    float t0 = x0 + dot * INV_S3;

    float cx = x2 * dz - x3 * dy;
    float cy = x3 * dx - x1 * dz;
    float cz = x1 * dy - x2 * dx;

    float t1 = x0 * dx + x1 + cx * INV_S2;
    float t2 = x0 * dy + x2 + cy * INV_S2;
    float t3 = x0 * dz + x3 + cz * INV_S2;

    float* o = out + ((size_t)r * 16 + f) * 4;
    atom_add_f32(o + 0, t0);
    atom_add_f32(o + 1, t1);
    atom_add_f32(o + 2, t2);
    atom_add_f32(o + 3, t3);
}

// ---------------------------------------------------------------------------
// Kernel 3: in-place rows(16) x 4 @ W(4x4) + b via V_WMMA_F32_16X16X4_F32.
//   A (16x4 f32): lanes 0-15 hold M=0..15 K={0,1}; lanes 16-31 hold K={2,3}.
//   B (4x16 f32): lane n (0-15) holds column n, K={0,1}; lanes 16-31 K={2,3}.
//     Only columns n<4 carry W; rest zero.
//   D (16x16 f32, 8 VGPRs): lane group (lane>=16 -> M+=8), N=lane&15.
// Each wave transforms 16 rows in place (row-local, no cross-row deps).
// ---------------------------------------------------------------------------
__global__ void gemm_wmma_kernel(float* __restrict__ io,
                                 const float* __restrict__ W,
                                 const float* __restrict__ bias,
                                 int rows) {
    int lane   = threadIdx.x & 31;
    int waveId = threadIdx.x >> 5;
    int rowBase = blockIdx.x * 128 + waveId * 16;
    if (rowBase + 16 > rows) return;   // wave-uniform -> EXEC stays all-ones

    int half = lane >> 4;   // 0: K=0,1  1: K=2,3
    int n    = lane & 15;   // A: M index; B/D: N index
    int row  = rowBase + n;

    // A: agg row, pre-scaled by 1/DENOM
    const float* ap = io + (size_t)row * 4 + half * 2;
    v2f a;
    a.x = ap[0] * 0.03125f;
    a.y = ap[1] * 0.03125f;

    // B: W padded to 4x16 (zero for n>=4)
    int k0 = half * 2;
    v2f bm;
    bm.x = (n < 4) ? W[k0 * 4 + n]       : 0.0f;
    bm.y = (n < 4) ? W[(k0 + 1) * 4 + n] : 0.0f;

    v8f c = {};
    v8f d = __builtin_amdgcn_wmma_f32_16x16x4_f32(
        /*neg_a=*/false, a, /*neg_b=*/false, bm,
        /*c_mod=*/(short)0, c, /*reuse_a=*/false, /*reuse_b=*/false);

    if (n < 4) {
        float bb = bias[n];
        int mBase = half * 8;
        #pragma unroll
        for (int j = 0; j < 8; ++j) {
            io[(size_t)(rowBase + mBase + j) * 4 + n] = d[j] + bb;
        }
    }
}

// ---------------------------------------------------------------------------
extern "C" void kernel_launch(void* const* d_in, const int* in_sizes, int n_in,
                              void* d_out, int out_size, void* d_ws, size_t ws_size,
                              hipStream_t stream) {
    const float* pos   = (const float*)d_in[0];  // (N,3)
    const float* feats = (const float*)d_in[1];  // (N,16,4)
    const float* W     = (const float*)d_in[2];  // (4,4)
    const float* bias  = (const float*)d_in[3];  // (4,)
    const int*   snd   = (const int*)d_in[4];    // (E,)
    const int*   rcv   = (const int*)d_in[5];    // (E,)
    float*       out   = (float*)d_out;          // (N,16,4)

    int N = in_sizes[0] / 3;
    int E = in_sizes[4];
    int total = N * 16 * 4;        // == out_size

    // 1) zero accumulator
    int n4 = total / 4;
    zero_f4_kernel<<<(n4 + 255) / 256, 256, 0, stream>>>((float4*)out, n4);

    // 2) edge tensor-product + scatter-add (one thread per edge-feature)
    long threads = (long)E * 16;
    int blocks = (int)((threads + 255) / 256);
    edge_tp_scatter_kernel<<<blocks, 256, 0, stream>>>(pos, feats, snd, rcv, out, E);

    // 3) in-place (rows x 4) @ W + b with WMMA; 128 rows per 256-thread block
    int rows = N * 16;
    gemm_wmma_kernel<<<(rows + 127) / 128, 256, 0, stream>>>(out, W, bias, rows);
}